// incidenceGuesser_20435454395056
// MI455X (gfx1250) — compile-verified
//
#include <hip/hip_runtime.h>

typedef __attribute__((ext_vector_type(2))) float v2f;
typedef __attribute__((ext_vector_type(8))) float v8f;

#define T_LEN     131072
#define IN_DIM    60
#define HID       14
#define G4        56      // 4*H
#define OUT_DIM   7
#define XW_STRIDE 64      // padded row stride for xw0 (unconditional stores)

__device__ __forceinline__ float sigf(float x) {
    return 1.0f / (1.0f + __expf(-x));
}

// ---------------------------------------------------------------------------
// Kernel 1: xw0 = x @ W_ih0^T + b_ih0   [T,60] x [60,56] -> [T,64] (padded)
// One wave per 16x16 output tile, fp32 WMMA, K=60 as 15 steps of K=4.
// Column index is CLAMPED (not predicated) for B/bias loads: garbage columns
// 56..63 are computed but harmless (WMMA columns independent); stores are
// unconditional into the padded stride-64 buffer. No exec-mask branching.
// ---------------------------------------------------------------------------
__global__ __launch_bounds__(256) void xproj_kernel(const float* __restrict__ x,
                                                    const float* __restrict__ Wih0,
                                                    const float* __restrict__ bih0,
                                                    float* __restrict__ xw0) {
    const int wave  = (blockIdx.x * blockDim.x + threadIdx.x) >> 5;
    const int lane  = threadIdx.x & 31;
    const int ntile = wave & 3;        // 4 N-tiles cover 64 cols (56 valid)
    const int mtile = wave >> 2;       // 8192 M-tiles cover T
    const int row   = mtile * 16 + (lane & 15);
    const int col   = ntile * 16 + (lane & 15);
    const int col_c = (col < G4) ? col : (G4 - 1);   // clamp -> always in-bounds
    const int kb    = (lane >> 4) * 2;               // K sub-offset per half-wave

    const float bias = bih0[col_c];
    v8f acc;
#pragma unroll
    for (int r = 0; r < 8; ++r) acc[r] = bias;       // C starts at broadcast bias

#pragma unroll
    for (int k = 0; k < IN_DIM; k += 4) {            // 15 x V_WMMA_F32_16X16X4_F32
        v2f a, b;
        a.x = x[(size_t)row * IN_DIM + k + kb];
        a.y = x[(size_t)row * IN_DIM + k + kb + 1];
        b.x = Wih0[(size_t)col_c * IN_DIM + k + kb];
        b.y = Wih0[(size_t)col_c * IN_DIM + k + kb + 1];
        acc = __builtin_amdgcn_wmma_f32_16x16x4_f32(false, a, false, b,
                                                    (short)0, acc, false, false);
    }

    const int rbase = mtile * 16 + ((lane >> 4) * 8);
#pragma unroll
    for (int r = 0; r < 8; ++r) {
        xw0[(size_t)(rbase + r) * XW_STRIDE + col] = acc[r];   // unconditional
    }
}

// ---------------------------------------------------------------------------
// Kernel 2: the serial LSTM scan. One workgroup, 3 waves = 3 layers,
// software-pipelined in time: wave l handles timestep t = s - l at superstep s.
// h forwarded layer->layer through double-buffered LDS; weights + h/c state in
// VGPRs; 14-wide dots via wave32 shuffles. One barrier per superstep.
// Lanes 28..31 compute garbage gates that no shuffle ever consumes -> no
// per-lane guards in the hot loop.
// ---------------------------------------------------------------------------
__global__ __launch_bounds__(96) void lstm_scan_kernel(
    const float* __restrict__ xw0,
    const float* __restrict__ h0,   const float* __restrict__ c0,
    const float* __restrict__ Whh0, const float* __restrict__ bhh0,
    const float* __restrict__ Wih1, const float* __restrict__ Whh1,
    const float* __restrict__ bih1, const float* __restrict__ bhh1,
    const float* __restrict__ Wih2, const float* __restrict__ Whh2,
    const float* __restrict__ bih2, const float* __restrict__ bhh2,
    float* __restrict__ h2out) {

    __shared__ float hbuf[2][2][16];   // [parity][link 0:L0->L1, 1:L1->L2][j]

    const int wave = threadIdx.x >> 5;   // 0,1,2 = layer
    const int lane = threadIdx.x & 31;
    const int g0 = lane;                          // always < 56
    const int g1c = (lane + 28 < G4) ? (lane + 28) : (G4 - 1);  // clamped

    // per-layer weight pointers (wave 0's Wih/bih values are never consumed)
    const float* Wih = (wave == 2) ? Wih2 : Wih1;
    const float* Whh = (wave == 0) ? Whh0 : (wave == 1) ? Whh1 : Whh2;
    const float* bih = (wave == 2) ? bih2 : bih1;
    const float* bhh = (wave == 0) ? bhh0 : (wave == 1) ? bhh1 : bhh2;

    float wih_a[HID], wih_b[HID], whh_a[HID], whh_b[HID];
#pragma unroll
    for (int k = 0; k < HID; ++k) {
        whh_a[k] = Whh[g0 * HID + k];
        whh_b[k] = Whh[g1c * HID + k];
        wih_a[k] = (wave != 0) ? Wih[g0 * HID + k] : 0.0f;
        wih_b[k] = (wave != 0) ? Wih[g1c * HID + k] : 0.0f;
    }
    const float bias_a = bhh[g0]  + ((wave != 0) ? bih[g0]  : 0.0f);
    const float bias_b = bhh[g1c] + ((wave != 0) ? bih[g1c] : 0.0f);

    float h_self = (lane < HID) ? h0[wave * HID + lane] : 0.0f;
    float c_self = (lane < HID) ? c0[wave * HID + lane] : 0.0f;

    if (threadIdx.x < 64) ((float*)hbuf)[threadIdx.x] = 0.0f;
    __syncthreads();

    for (int s = 0; s < T_LEN + 2; ++s) {
        const int t = s - wave;
        const bool active = (t >= 0) && (t < T_LEN);   // wave-uniform

        float hin  = 0.0f;
        float acc0 = bias_a, acc1 = bias_b;

        if (active) {
            if (wave == 0) {
                const float* xp = xw0 + (size_t)t * XW_STRIDE;
                acc0 += xp[lane];            // cols 0..31 (28..31 unused garbage)
                acc1 += xp[lane + 28];       // cols 28..59 (56..59 = pad)
                __builtin_prefetch(xp + 16 * XW_STRIDE, 0, 0);  // stream ahead
            } else {
                hin = hbuf[(s + 1) & 1][wave - 1][lane & 15];
            }
        }

        if (wave != 0) {           // inter-layer projection h_{l-1}(t) @ Wih^T
#pragma unroll
            for (int k = 0; k < HID; ++k) {
                const float hk = __shfl(hin, k, 32);
                acc0 = fmaf(hk, wih_a[k], acc0);
                acc1 = fmaf(hk, wih_b[k], acc1);
            }
        }
#pragma unroll
        for (int k = 0; k < HID; ++k) {   // recurrent h_l(t-1) @ Whh^T
            const float hk = __shfl(h_self, k, 32);
            acc0 = fmaf(hk, whh_a[k], acc0);
            acc1 = fmaf(hk, whh_b[k], acc1);
        }

        // gate gather: unit j: i=g[j](lane j,acc0) f=g[j+14](lane j+14,acc0)
        //              g=g[j+28](lane j,acc1)      o=g[j+42](lane j+14,acc1)
        const int src = (lane < HID) ? (lane + HID) : lane;
        const float f_g = __shfl(acc0, src, 32);
        const float o_g = __shfl(acc1, src, 32);

        if (active && lane < HID) {
            const float cn = sigf(f_g) * c_self + sigf(acc0) * tanhf(acc1);
            const float hn = sigf(o_g) * tanhf(cn);
            c_self = cn;
            h_self = hn;
            if (wave < 2) hbuf[s & 1][wave][lane] = hn;
            else          h2out[(size_t)t * HID + lane] = hn;
        }
        __syncthreads();
    }
}

// ---------------------------------------------------------------------------
// Kernel 3: out = relu(relu(h2) @ W_lin^T + b_lin)   [T,14] x [14,7] -> [T,7]
// One wave per 16-row tile; K padded 14->16 (4 WMMA steps), N padded 7->16.
// K-tail handled with clamped addresses (keeps b64 pairs) + value cndmask.
// ---------------------------------------------------------------------------
__global__ __launch_bounds__(256) void head_kernel(const float* __restrict__ h2,
                                                   const float* __restrict__ Wlin,
                                                   const float* __restrict__ blin,
                                                   float* __restrict__ out) {
    const int wave = (blockIdx.x * blockDim.x + threadIdx.x) >> 5;
    const int lane = threadIdx.x & 31;
    const int row  = wave * 16 + (lane & 15);
    const int col  = lane & 15;
    const int col_c = (col < OUT_DIM) ? col : (OUT_DIM - 1);   // clamp
    const int kb   = (lane >> 4) * 2;
    const bool colv = col < OUT_DIM;

    const float bias = blin[col_c];
    v8f acc;
#pragma unroll
    for (int r = 0; r < 8; ++r) acc[r] = bias;

#pragma unroll
    for (int k = 0; k < 16; k += 4) {
        const int k0 = k + kb;
        const int k1 = k0 + 1;
        const int kc = (k0 < HID - 2) ? k0 : (HID - 2);  // clamp, keeps pair adjacent
        // unconditional in-bounds loads (pairs stay consecutive -> b64)
        const float a0 = h2[(size_t)row * HID + kc];
        const float a1 = h2[(size_t)row * HID + kc + 1];
        const float b0 = Wlin[col_c * HID + kc];
        const float b1 = Wlin[col_c * HID + kc + 1];
        v2f a, b;
        a.x = (k0 < HID) ? fmaxf(a0, 0.0f) : 0.0f;   // v_cndmask, no branch
        a.y = (k1 < HID) ? fmaxf(a1, 0.0f) : 0.0f;
        b.x = (k0 < HID) ? b0 : 0.0f;
        b.y = (k1 < HID) ? b1 : 0.0f;
        acc = __builtin_amdgcn_wmma_f32_16x16x4_f32(false, a, false, b,
                                                    (short)0, acc, false, false);
    }

    const int rbase = wave * 16 + ((lane >> 4) * 8);
    if (colv) {                      // single uniform-per-lane guard
#pragma unroll
        for (int r = 0; r < 8; ++r) {
            out[(size_t)(rbase + r) * OUT_DIM + col] = fmaxf(acc[r], 0.0f);
        }
    }
}

// ---------------------------------------------------------------------------
extern "C" void kernel_launch(void* const* d_in, const int* in_sizes, int n_in,
                              void* d_out, int out_size, void* d_ws, size_t ws_size,
                              hipStream_t stream) {
    (void)in_sizes; (void)n_in; (void)out_size; (void)ws_size;

    const float* x    = (const float*)d_in[0];
    const float* h0   = (const float*)d_in[1];
    const float* c0   = (const float*)d_in[2];
    const float* Wih0 = (const float*)d_in[3];
    const float* Whh0 = (const float*)d_in[4];
    const float* bih0 = (const float*)d_in[5];
    const float* bhh0 = (const float*)d_in[6];
    const float* Wih1 = (const float*)d_in[7];
    const float* Whh1 = (const float*)d_in[8];
    const float* bih1 = (const float*)d_in[9];
    const float* bhh1 = (const float*)d_in[10];
    const float* Wih2 = (const float*)d_in[11];
    const float* Whh2 = (const float*)d_in[12];
    const float* bih2 = (const float*)d_in[13];
    const float* bhh2 = (const float*)d_in[14];
    const float* Wlin = (const float*)d_in[15];
    const float* blin = (const float*)d_in[16];
    float* out = (float*)d_out;

    float* xw0 = (float*)d_ws;                         // [T,64] ~33.6 MB (padded)
    float* h2  = xw0 + (size_t)T_LEN * XW_STRIDE;      // [T,14] ~ 7.3 MB

    // 8192 M-tiles x 4 N-tiles = 32768 waves, 8 waves per 256-thread block
    xproj_kernel<<<4096, 256, 0, stream>>>(x, Wih0, bih0, xw0);

    // serial scan: one workgroup, 3 waves (one per layer), pipelined in time
    lstm_scan_kernel<<<1, 96, 0, stream>>>(xw0, h0, c0,
                                           Whh0, bhh0,
                                           Wih1, Whh1, bih1, bhh1,
                                           Wih2, Whh2, bih2, bhh2, h2);

    // 8192 waves, 8 per block
    head_kernel<<<1024, 256, 0, stream>>>(h2, Wlin, blin, out);
}